// Chebyshev2_34514357191197
// MI455X (gfx1250) — compile-verified
//
#include <hip/hip_runtime.h>

// ---------------------------------------------------------------------------
// ChebNet conv, K=4, N=4, Fin=32, Fout=64, M=100000, E=1.6M.
//
// Phase 1 (sparse, L2-resident): T0 = transpose(x); then 3x:
//   T_k = 2*A*T_{k-1} - 2*T_{k-1} - T_{k-2}   (T1 = A*T0 - T0)
// done as an elementwise init (axpby) + wave-per-edge atomic scatter
// (global_atomic_add_f32; 51.2 MB accumulator lives in the 192 MB L2).
//
// Phase 2 (dense): out[r, :] = sum_{f,k} T_k[m*128 + 4f + n] * W[4f+k, :]
// with r = n*M + m, using V_WMMA_F32_16X16X4_F32 (fp32-exact; workload is
// memory bound so low-precision WMMA buys nothing). W is staged in LDS in a
// pair-interleaved layout so every B fragment is one aligned ds_load_b64
// into an even VGPR pair (no re-pairing movs feeding the WMMAs).
//
// Workspace requirement: 4 * M * 128 * 4 bytes = 204.8 MB.
// ---------------------------------------------------------------------------

typedef __attribute__((ext_vector_type(2))) float v2f;
typedef __attribute__((ext_vector_type(8))) float v8f;

// T0[m*128 + f*4 + n] = x[n, m, f]   (x is [N=4, M, Fin=32])
__global__ __launch_bounds__(256) void cheb_build_T0(
    const float* __restrict__ x, float* __restrict__ T0, int M) {
  long long idx = (long long)blockIdx.x * 256 + threadIdx.x;
  long long total = (long long)M * 128;
  if (idx >= total) return;
  int nn = (int)(idx & 3);
  int f  = (int)((idx >> 2) & 31);
  long long m = idx >> 7;
  T0[idx] = x[((long long)nn * M + m) * 32 + f];
}

// out = a*X + b*Y (elementwise, float4-vectorized; n is a multiple of 4)
__global__ __launch_bounds__(256) void cheb_axpby4(
    float4* __restrict__ out, const float4* __restrict__ X,
    const float4* __restrict__ Y, float a, float b, int n4) {
  int i = blockIdx.x * 256 + threadIdx.x;
  if (i >= n4) return;
  float4 xv = X[i], yv = Y[i], r;
  r.x = a * xv.x + b * yv.x;
  r.y = a * xv.y + b * yv.y;
  r.z = a * xv.z + b * yv.z;
  r.w = a * xv.w + b * yv.w;
  out[i] = r;
}

// One wave per edge: Tout[row,:] += scale*val * Tin[col,:] (128 floats/row,
// 4 floats per lane, f32 global atomics resolving in L2).
__global__ __launch_bounds__(256) void cheb_scatter_edges(
    const float* __restrict__ Tin, float* __restrict__ Tout,
    const float* __restrict__ vals, const int* __restrict__ rows,
    const int* __restrict__ cols, int E, float scale) {
  long long gid = (long long)blockIdx.x * 256 + threadIdx.x;
  int e = (int)(gid >> 5);
  if (e >= E) return;
  int lane = (int)(gid & 31);
  int r = rows[e];
  int c = cols[e];
  float v = vals[e] * scale;
  const float4 s = *(const float4*)(Tin + (size_t)c * 128 + lane * 4);
  float* d = Tout + (size_t)r * 128 + lane * 4;
  unsafeAtomicAdd(d + 0, v * s.x);
  unsafeAtomicAdd(d + 1, v * s.y);
  unsafeAtomicAdd(d + 2, v * s.z);
  unsafeAtomicAdd(d + 3, v * s.w);
}

// Dense stage: out[n*M+m, o] = sum_{f=0..31, k=0..3} Tk[m*128+4f+n] * W[4f+k, o]
// 8 waves/block, one 16-row tile per wave, 4 N-tiles (Fout=64) per wave.
// A-fragment layout for V_WMMA_F32_16X16X4_F32: lane l, VGPR j holds
// A[M=l%16, K=2*(l/16)+j]  ->  lane-half selects T buffer pair (T0,T1)/(T2,T3).
// B-fragment mirrors: lane l, VGPR j holds B[K=2*(l/16)+j, N=l%16].
// C/D: lane l, VGPR v holds D[M = v + 8*(l/16), N = l%16].
//
// LDS W layout (pair-interleaved): sW2[(p*64 + o)*2 + j] = W[(2p+j)*64 + o],
// p = 0..63 (K-row pair), so a B fragment is one 8-byte ds_load_b64 and the
// four N-tiles sit at immediate offsets +0/+128/+256/+384 bytes.
__global__ __launch_bounds__(256) void cheb_gemm_wmma(
    const float* __restrict__ T0, const float* __restrict__ T1,
    const float* __restrict__ T2, const float* __restrict__ T3,
    const float* __restrict__ W, float* __restrict__ out,
    int M, int numTiles) {
  __shared__ float sW2[128 * 64];  // 32 KB of the 320 KB/WGP LDS
  // Cooperative pair-interleaved fill: two coalesced global streams,
  // contiguous b64 LDS stores.
  for (int i = threadIdx.x; i < 64 * 64; i += 256) {
    const int p = i >> 6;        // K-row pair index
    const int o = i & 63;        // output column
    v2f w;
    w.x = W[(2 * p + 0) * 64 + o];
    w.y = W[(2 * p + 1) * 64 + o];
    *(v2f*)&sW2[(size_t)i * 2] = w;
  }
  __syncthreads();

  const int lane = threadIdx.x & 31;
  const int wave = threadIdx.x >> 5;
  const int tile = blockIdx.x * 8 + wave;
  if (tile >= numTiles) return;  // wave-uniform: EXEC stays all-1s for WMMA

  const int r0 = tile * 16;
  const int n  = r0 / M;        // all 16 rows share n (M % 16 == 0)
  const int m0 = r0 - n * M;

  const int half = lane >> 4;   // 0: lanes 0-15, 1: lanes 16-31
  const int lrow = lane & 15;
  const float* __restrict__ TA = half ? T2 : T0;  // k = 2*half
  const float* __restrict__ TB = half ? T3 : T1;  // k = 2*half + 1
  const size_t aoff = (size_t)(m0 + lrow) * 128 + n;

  v8f acc0 = {}, acc1 = {}, acc2 = {}, acc3 = {};

#pragma unroll 4
  for (int f = 0; f < 32; ++f) {
    v2f a;
    a.x = TA[aoff + 4 * f];
    a.y = TB[aoff + 4 * f];
    // B fragments: one aligned b64 LDS load each, immediate offsets per N-tile.
    const v2f* bp = (const v2f*)&sW2[(size_t)(((f << 1) + half) * 64 + lrow) * 2];
    const v2f b0 = bp[0];
    const v2f b1 = bp[16];
    const v2f b2 = bp[32];
    const v2f b3 = bp[48];
    acc0 = __builtin_amdgcn_wmma_f32_16x16x4_f32(false, a, false, b0, (short)0, acc0, false, false);
    acc1 = __builtin_amdgcn_wmma_f32_16x16x4_f32(false, a, false, b1, (short)0, acc1, false, false);
    acc2 = __builtin_amdgcn_wmma_f32_16x16x4_f32(false, a, false, b2, (short)0, acc2, false, false);
    acc3 = __builtin_amdgcn_wmma_f32_16x16x4_f32(false, a, false, b3, (short)0, acc3, false, false);
  }

  float* obase = out + (size_t)r0 * 64 + lrow;
#pragma unroll
  for (int v = 0; v < 8; ++v) {
    const int mr = v + 8 * half;
    float* o = obase + (size_t)mr * 64;
    o[0]  = acc0[v];
    o[16] = acc1[v];
    o[32] = acc2[v];
    o[48] = acc3[v];
  }
}

extern "C" void kernel_launch(void* const* d_in, const int* in_sizes, int n_in,
                              void* d_out, int out_size, void* d_ws, size_t ws_size,
                              hipStream_t stream) {
  const float* x         = (const float*)d_in[0];
  const float* edge_vals = (const float*)d_in[1];
  const float* W         = (const float*)d_in[2];
  const int*   edge_rows = (const int*)d_in[3];
  const int*   edge_cols = (const int*)d_in[4];

  const int N = 4, Fin = 32;          // fixed by the reference
  const int E = in_sizes[1];
  const int M = in_sizes[0] / (N * Fin);
  const size_t rowElems = (size_t)M * 128;      // Fin*N = 128 floats per vertex

  if (ws_size < 4 * rowElems * sizeof(float)) return;  // need 204.8 MB scratch

  float* T0 = (float*)d_ws;
  float* T1 = T0 + rowElems;
  float* T2 = T1 + rowElems;
  float* T3 = T2 + rowElems;

  const int eltBlocks  = (int)((rowElems + 255) / 256);
  const int vec4Blocks = (int)((rowElems / 4 + 255) / 256);
  const int edgeBlocks = (int)(((long long)E * 32 + 255) / 256);

  // T0 = transpose(x)
  cheb_build_T0<<<eltBlocks, 256, 0, stream>>>(x, T0, M);

  // T1 = A*T0 - T0
  cheb_axpby4<<<vec4Blocks, 256, 0, stream>>>((float4*)T1, (const float4*)T0,
                                              (const float4*)T0, -1.f, 0.f,
                                              (int)(rowElems / 4));
  cheb_scatter_edges<<<edgeBlocks, 256, 0, stream>>>(T0, T1, edge_vals,
                                                     edge_rows, edge_cols, E, 1.f);

  // T2 = 2*A*T1 - 2*T1 - T0
  cheb_axpby4<<<vec4Blocks, 256, 0, stream>>>((float4*)T2, (const float4*)T1,
                                              (const float4*)T0, -2.f, -1.f,
                                              (int)(rowElems / 4));
  cheb_scatter_edges<<<edgeBlocks, 256, 0, stream>>>(T1, T2, edge_vals,
                                                     edge_rows, edge_cols, E, 2.f);

  // T3 = 2*A*T2 - 2*T2 - T1
  cheb_axpby4<<<vec4Blocks, 256, 0, stream>>>((float4*)T3, (const float4*)T2,
                                              (const float4*)T1, -2.f, -1.f,
                                              (int)(rowElems / 4));
  cheb_scatter_edges<<<edgeBlocks, 256, 0, stream>>>(T2, T3, edge_vals,
                                                     edge_rows, edge_cols, E, 2.f);

  // Dense WMMA stage: out = A_eff @ W
  const int numTiles = (int)(((long long)N * M) / 16);  // M % 16 == 0
  const int gemmBlocks = (numTiles + 7) / 8;
  cheb_gemm_wmma<<<gemmBlocks, 256, 0, stream>>>(T0, T1, T2, T3, W,
                                                 (float*)d_out, M, numTiles);
}